// Q_Act_d_13176959664397
// MI455X (gfx1250) — compile-verified
//
#include <hip/hip_runtime.h>
#include <math.h>

// Problem shape (fixed by the reference)
#define BB   16
#define LL   4000
#define DD   1024
#define NR   10
#define NBK  (LL / NR)      // 400
#define NG   (BB * NBK)     // 6400 row-groups
#define TPB  256            // 8 wave32s; each thread owns 4 elems of D
#define QMAXV 255.0f
#define INV_QRANGE (1.0f / 255.0f)

#if __has_builtin(__builtin_amdgcn_global_load_async_to_lds_b128) && \
    __has_builtin(__builtin_amdgcn_s_wait_asynccnt)
#define USE_ASYNC 1
#endif

typedef int   v4i __attribute__((vector_size(16)));
typedef float v4f __attribute__((ext_vector_type(4)));
typedef __attribute__((address_space(1))) v4i* as1_v4i_ptr;
typedef __attribute__((address_space(3))) v4i* as3_v4i_ptr;

// ---------------- wave32-aware block reductions (8 waves / block) -----------

__device__ __forceinline__ void block_sum2(float& a, float& b, float* red) {
  #pragma unroll
  for (int off = 16; off > 0; off >>= 1) {
    a += __shfl_xor(a, off, 32);
    b += __shfl_xor(b, off, 32);
  }
  const int w = threadIdx.x >> 5, lane = threadIdx.x & 31;
  __syncthreads();                 // protect LDS buffer reuse
  if (lane == 0) { red[w] = a; red[8 + w] = b; }
  __syncthreads();
  float sa = 0.f, sb = 0.f;
  #pragma unroll
  for (int j = 0; j < 8; ++j) { sa += red[j]; sb += red[8 + j]; }
  a = sa; b = sb;                  // broadcast to all threads
}

__device__ __forceinline__ void block_maxmin(float& mx, float& mn, float* red) {
  #pragma unroll
  for (int off = 16; off > 0; off >>= 1) {
    mx = fmaxf(mx, __shfl_xor(mx, off, 32));
    mn = fminf(mn, __shfl_xor(mn, off, 32));
  }
  const int w = threadIdx.x >> 5, lane = threadIdx.x & 31;
  __syncthreads();
  if (lane == 0) { red[w] = mx; red[8 + w] = mn; }
  __syncthreads();
  float a = -3.402823466e38f, b = 3.402823466e38f;
  #pragma unroll
  for (int j = 0; j < 8; ++j) { a = fmaxf(a, red[j]); b = fminf(b, red[8 + j]); }
  mx = a; mn = b;
}

// ---------------- kernels ---------------------------------------------------

__global__ void zero_flags_k(unsigned int* flags) {
  if (threadIdx.x < 16) flags[threadIdx.x] = 0u;
}

// cond flag for t = 0 (o_list == 0, so inlier == x row 0 of each group)
__global__ __launch_bounds__(TPB) void preflag_k(const float* __restrict__ x,
                                                 const float* __restrict__ ss,
                                                 unsigned int* flags) {
  __shared__ float red[16];
  const int g = blockIdx.x, b = g / NBK, blk = g - b * NBK;
  const int row = blk * NR;
  const float4 v = ((const float4*)(x + ((size_t)b * LL + row) * DD))[threadIdx.x];
  float mx = fmaxf(fmaxf(v.x, v.y), fmaxf(v.z, v.w));
  float mn = fminf(fminf(v.x, v.y), fminf(v.z, v.w));
  block_maxmin(mx, mn, red);
  if (threadIdx.x == 0) {
    if ((mx - mn) * INV_QRANGE > ss[row]) atomicOr(&flags[0], 1u);
  }
}

__global__ __launch_bounds__(TPB) void step_k(const float* __restrict__ x,
                                              const float* __restrict__ ss,
                                              const float* __restrict__ sz,
                                              float* __restrict__ y,
                                              unsigned char* __restrict__ o_ws,
                                              unsigned int* flags, int t) {
  __shared__ float red[16];
  __shared__ float4 stage[TPB];   // async-staged next row (4 KB)
  const int tid = threadIdx.x;
  const int g = blockIdx.x, b = g / NBK, blk = g - b * NBK;
  const int row = blk * NR + t;
  const size_t xoff = ((size_t)b * LL + row) * (size_t)DD;

#ifdef USE_ASYNC
  // Prefetch next row into LDS via the gfx1250 async DMA path; each lane
  // copies its own 16B slice so only s_wait_asynccnt (per-wave) is needed.
  if (t < NR - 1) {
    __builtin_amdgcn_global_load_async_to_lds_b128(
        (as1_v4i_ptr)(x + xoff + DD + (size_t)tid * 4),
        (as3_v4i_ptr)&stage[tid],
        0, 0);
  }
#endif

  const float4 xv = ((const float4*)(x + xoff))[tid];

  // outlier mask, packed as 1 byte per element (L2-resident, 6.5 MB total)
  unsigned int* op = (unsigned int*)(o_ws + (size_t)g * DD) + tid;
  float o0, o1, o2, o3;
  if (t == 0) {
    o0 = o1 = o2 = o3 = 0.f;
  } else {
    const unsigned int ob = *op;
    o0 = (float)(ob & 0xffu);
    o1 = (float)((ob >> 8) & 0xffu);
    o2 = (float)((ob >> 16) & 0xffu);
    o3 = (float)(ob >> 24);
  }
  float i0 = xv.x * (1.f - o0), i1 = xv.y * (1.f - o1);
  float i2 = xv.z * (1.f - o2), i3 = xv.w * (1.f - o3);

  // cond (global any over all groups, computed by the previous launch)
  if (flags[t]) {
    const float a0 = fabsf(i0), a1 = fabsf(i1), a2 = fabsf(i2), a3 = fabsf(i3);
    float s1 = (a0 + a1) + (a2 + a3);
    float s2 = (a0 * a0 + a1 * a1) + (a2 * a2 + a3 * a3);
    block_sum2(s1, s2, red);
    const float mean = s1 * (1.0f / 1024.0f);
    const float var  = (s2 - s1 * mean) * (1.0f / 1023.0f);   // ddof = 1
    const float thr  = mean + 4.0f * sqrtf(fmaxf(var, 0.0f));
    o0 = (o0 != 0.f || a0 > thr) ? 1.f : 0.f;
    o1 = (o1 != 0.f || a1 > thr) ? 1.f : 0.f;
    o2 = (o2 != 0.f || a2 > thr) ? 1.f : 0.f;
    o3 = (o3 != 0.f || a3 > thr) ? 1.f : 0.f;
    i0 = xv.x * (1.f - o0); i1 = xv.y * (1.f - o1);
    i2 = xv.z * (1.f - o2); i3 = xv.w * (1.f - o3);
  }
  *op = (unsigned int)o0 | ((unsigned int)o1 << 8) |
        ((unsigned int)o2 << 16) | ((unsigned int)o3 << 24);

  // outlier branch: out = x * o, dynamic 8-bit requant from row max/min
  const float u0 = xv.x * o0, u1 = xv.y * o1, u2 = xv.z * o2, u3 = xv.w * o3;
  float mx = fmaxf(fmaxf(u0, u1), fmaxf(u2, u3));
  float mn = fminf(fminf(u0, u1), fminf(u2, u3));
  block_maxmin(mx, mn, red);
  const float o_s = fmaxf(mx - mn, 1e-8f) * INV_QRANGE;
  const float o_z = rintf(-mn / o_s);

  const float s = ss[row], z = sz[row];

  v4f out;
  {
    float q, dq, oq, od;
    q  = fminf(fmaxf(rintf(i0 / s) + z, 0.f), QMAXV);  dq = (q - z) * s;
    oq = fminf(fmaxf(rintf(u0 / o_s) + o_z, 0.f), QMAXV); od = (oq - o_z) * o_s;
    out.x = dq + od;
    q  = fminf(fmaxf(rintf(i1 / s) + z, 0.f), QMAXV);  dq = (q - z) * s;
    oq = fminf(fmaxf(rintf(u1 / o_s) + o_z, 0.f), QMAXV); od = (oq - o_z) * o_s;
    out.y = dq + od;
    q  = fminf(fmaxf(rintf(i2 / s) + z, 0.f), QMAXV);  dq = (q - z) * s;
    oq = fminf(fmaxf(rintf(u2 / o_s) + o_z, 0.f), QMAXV); od = (oq - o_z) * o_s;
    out.z = dq + od;
    q  = fminf(fmaxf(rintf(i3 / s) + z, 0.f), QMAXV);  dq = (q - z) * s;
    oq = fminf(fmaxf(rintf(u3 / o_s) + o_z, 0.f), QMAXV); od = (oq - o_z) * o_s;
    out.w = dq + od;
  }
  // streamed output: non-temporal so it doesn't evict the L2 working set
  __builtin_nontemporal_store(out, (v4f*)(y + xoff) + tid);

  // produce the global cond flag for step t+1 using the updated mask and the
  // async-staged next row
  if (t < NR - 1) {
#ifdef USE_ASYNC
    __builtin_amdgcn_s_wait_asynccnt(0);
    const float4 xn = stage[tid];
#else
    const float4 xn = ((const float4*)(x + xoff + DD))[tid];
#endif
    const float n0 = xn.x * (1.f - o0), n1 = xn.y * (1.f - o1);
    const float n2 = xn.z * (1.f - o2), n3 = xn.w * (1.f - o3);
    float nmx = fmaxf(fmaxf(n0, n1), fmaxf(n2, n3));
    float nmn = fminf(fminf(n0, n1), fminf(n2, n3));
    block_maxmin(nmx, nmn, red);
    if (tid == 0) {
      if ((nmx - nmn) * INV_QRANGE > ss[row + 1]) atomicOr(&flags[t + 1], 1u);
    }
  }
}

// ---------------- host launch ----------------------------------------------

extern "C" void kernel_launch(void* const* d_in, const int* in_sizes, int n_in,
                              void* d_out, int out_size, void* d_ws, size_t ws_size,
                              hipStream_t stream) {
  (void)in_sizes; (void)n_in; (void)out_size; (void)ws_size;
  const float* x  = (const float*)d_in[0];
  const float* ss = (const float*)d_in[1];
  const float* sz = (const float*)d_in[2];
  float* y = (float*)d_out;

  unsigned int*  flags = (unsigned int*)d_ws;                 // 10 step flags
  unsigned char* o_ws  = (unsigned char*)d_ws + 1024;         // 6.25 MB mask

  zero_flags_k<<<1, 32, 0, stream>>>(flags);
  preflag_k<<<NG, TPB, 0, stream>>>(x, ss, flags);
  for (int t = 0; t < NR; ++t) {
    step_k<<<NG, TPB, 0, stream>>>(x, ss, sz, y, o_ws, flags, t);
  }
}